// net_38895223832615
// MI455X (gfx1250) — compile-verified
//
#include <hip/hip_runtime.h>
#include <hip/hip_bf16.h>
#include <math.h>

// ---------------------------------------------------------------------------
// MI455X (gfx1250) implementation.
//
// RoIAlign is memory bound (≈100MB stores + ≈400MB L2-resident gathers ->
// ~20us at 23.3 TB/s). We exploit bilinear separability so the y-direction
// interpolation becomes a real GEMM on the wave32 WMMA pipe:
//   pooled[iy, ix, c] = sum_a Wy[iy,a] * T[a, c*7+ix]   (M=16,K=16,N=896)
// computed as 4 k-steps of V_WMMA_F32_16X16X4_F32 per 16-wide N tile.
// Stage-1 (gather + x-interp into LDS) issues exactly as many global loads
// as direct RoIAlign (14*14 == 4*49 samples per channel plane).
// ---------------------------------------------------------------------------

typedef __attribute__((ext_vector_type(2))) float v2f;
typedef __attribute__((ext_vector_type(8))) float v8f;

#define N_ROIS   4000
#define CCH      128
#define NPOS     1000
#define NNEG     200000
#define POOL_OUT (N_ROIS * CCH * 49)   // 25,088,000
#define NB       16384                 // histogram bins for SHEM selection
#define NEG_BLOCKS ((NNEG + 255) / 256)

// ws float-index layout
#define WS_HIST   0            // NB ints
#define WS_CUT    (NB)         // 1 int
#define WS_POS    (NB + 1)     // 1 float (pos loss mean)
#define WS_NSUM   (NB + 16)    // NEG_BLOCKS floats
#define WS_NCNT   (NB + 16 + NEG_BLOCKS + 16)  // NEG_BLOCKS floats

// ---------------------------------------------------------------------------
// RoIAlign: one workgroup (256 threads = 8 waves) per ROI.
// ---------------------------------------------------------------------------
__global__ __launch_bounds__(256) void roialign_wmma_kernel(
    const float* __restrict__ f0, const float* __restrict__ f1,
    const float* __restrict__ f2, const float* __restrict__ f3,
    const float* __restrict__ rois, float* __restrict__ out)
{
    __shared__ float T[16 * 900];     // [y-slot][n = c*7 + ix], padded stride
    __shared__ float wyA[8], wxA[8];
    __shared__ int   ya[16], xs0[8], xs1[8];
    __shared__ int   sc[4];           // lvl, batch, H

    const int r   = blockIdx.x;
    const int tid = threadIdx.x;

    if (tid == 0) {
        float b0 = rois[r * 5 + 0], b1 = rois[r * 5 + 1];
        float b2 = rois[r * 5 + 2], b3 = rois[r * 5 + 3];
        int   batch = (int)rois[r * 5 + 4];
        float hh = b2 - b0, ww = b3 - b1;
        int lvl = (int)rintf(4.0f + log2f(sqrtf(hh * ww + 1e-12f)));
        lvl = lvl < 0 ? 0 : (lvl > 3 ? 3 : lvl);
        int H = 256 >> lvl;
        float y1 = b0 * (float)H, x1 = b1 * (float)H;
        float y2 = b2 * (float)H, x2 = b3 * (float)H;
        float bh = (y2 - y1) * (1.0f / 7.0f);
        float bw = (x2 - x1) * (1.0f / 7.0f);
        for (int i = 0; i < 7; ++i) {
            float sy = fminf(fmaxf(y1 + bh * ((float)i + 0.5f), 0.0f), (float)H - 1.0f);
            float y0f = floorf(sy);
            int   y0  = (int)y0f;
            ya[2 * i]     = y0;
            ya[2 * i + 1] = min(y0 + 1, H - 1);
            wyA[i] = sy - y0f;
            float sx = fminf(fmaxf(x1 + bw * ((float)i + 0.5f), 0.0f), (float)H - 1.0f);
            float x0f = floorf(sx);
            int   x0  = (int)x0f;
            xs0[i] = x0;
            xs1[i] = min(x0 + 1, H - 1);
            wxA[i] = sx - x0f;
        }
        ya[14] = 0; ya[15] = 0;
        sc[0] = lvl; sc[1] = batch; sc[2] = H;
    }
    __syncthreads();

    const int lvl = sc[0], batch = sc[1], H = sc[2];
    const float* F = (lvl == 0) ? f0 : (lvl == 1) ? f1 : (lvl == 2) ? f2 : f3;
    F += (size_t)batch * CCH * H * H;

    // Stage 1: gather + x-interp -> T[a][n], n = c*7 + ix. K rows 14,15 zero.
    for (int it = tid; it < 16 * 896; it += 256) {
        int a = it / 896;
        int n = it - a * 896;
        float v = 0.0f;
        if (a < 14) {
            int c  = n / 7;
            int ix = n - c * 7;
            const float* P = F + (size_t)c * H * H + (size_t)ya[a] * H;
            float w = wxA[ix];
            v = (1.0f - w) * P[xs0[ix]] + w * P[xs1[ix]];
        }
        T[a * 900 + n] = v;
    }
    __syncthreads();

    // Stage 2: pooled = Wy (16x16) x T (16x896) via v_wmma_f32_16x16x4_f32.
    const int lane  = tid & 31;
    const int wv    = tid >> 5;
    const int m     = lane & 15;     // A row / B column within tile
    const int khalf = lane >> 4;     // lanes 16..31 hold K+2 per ISA layout
    const float wym = (m < 7) ? wyA[m] : 0.0f;

    for (int tile = wv * 7; tile < wv * 7 + 7; ++tile) {
        v8f acc = {0.f, 0.f, 0.f, 0.f, 0.f, 0.f, 0.f, 0.f};
        const int nb = tile * 16 + m;
        #pragma unroll
        for (int ks = 0; ks < 4; ++ks) {
            const int k0 = 4 * ks + 2 * khalf;
            v2f A, Bv;
            // A = Wy[m, k]: nonzero only at k==2m (1-wy) and k==2m+1 (wy).
            A.x = (m < 7 && k0     == 2 * m) ? (1.0f - wym)
                : (m < 7 && k0     == 2 * m + 1) ? wym : 0.0f;
            A.y = (m < 7 && k0 + 1 == 2 * m) ? (1.0f - wym)
                : (m < 7 && k0 + 1 == 2 * m + 1) ? wym : 0.0f;
            Bv.x = T[(k0)     * 900 + nb];
            Bv.y = T[(k0 + 1) * 900 + nb];
            acc = __builtin_amdgcn_wmma_f32_16x16x4_f32(
                false, A, false, Bv, (short)0, acc, false, false);
        }
        // D layout: VGPR v -> rows v (lanes 0-15) and v+8 (lanes 16-31).
        // Only rows 0..6 (iy) are real output; lanes 16-31 hold rows >= 8.
        if (lane < 16) {
            int n  = tile * 16 + lane;
            int c  = n / 7;
            int ix = n - c * 7;
            float* O = out + (size_t)r * (CCH * 49) + c * 49 + ix;
            #pragma unroll
            for (int v = 0; v < 7; ++v) O[v * 7] = acc[v];
        }
    }
}

// ---------------------------------------------------------------------------
// Losses
// ---------------------------------------------------------------------------
__device__ __forceinline__ float block_reduce(float v, float* red) {
    int t = threadIdx.x;
    red[t] = v;
    __syncthreads();
    for (int s = 128; s > 0; s >>= 1) {
        if (t < s) red[t] += red[t + s];
        __syncthreads();
    }
    return red[0];
}

// single block: pos classification loss + smooth-L1 bbox loss
__global__ __launch_bounds__(256) void pos_bbox_kernel(
    const int* __restrict__ am, const float* __restrict__ logits,
    const float* __restrict__ pred, const float* __restrict__ targ,
    float* __restrict__ wsf, float* __restrict__ out)
{
    __shared__ float red[256];
    int t = threadIdx.x;
    float s = 0.0f;
    for (int i = t; i < NPOS; i += 256) {
        float l0 = logits[2 * i], l1 = logits[2 * i + 1];
        float mx = fmaxf(l0, l1);
        float lse = mx + logf(expf(l0 - mx) + expf(l1 - mx));
        float lt = (am[i] == 0) ? l0 : l1;
        s += lse - lt;   // -log_softmax[target]
    }
    float pos = block_reduce(s, red);
    __syncthreads();
    float sb = 0.0f;
    for (int i = t; i < NPOS * 4; i += 256) {
        float d = fabsf(pred[i] - targ[i]);
        sb += (d < 1.0f) ? 0.5f * d * d : d - 0.5f;
    }
    float bb = block_reduce(sb, red);
    if (t == 0) {
        wsf[WS_POS] = pos * (1.0f / (float)NPOS);
        out[POOL_OUT + 1] = bb * (1.0f / (float)(NPOS * 4));
    }
}

__global__ __launch_bounds__(256) void zero_hist_kernel(int* __restrict__ hist) {
    int i = blockIdx.x * 256 + threadIdx.x;
    if (i < NB) hist[i] = 0;
}

__global__ __launch_bounds__(256) void neg_hist_kernel(
    const float* __restrict__ logits, int* __restrict__ hist)
{
    int i = blockIdx.x * 256 + threadIdx.x;
    if (i >= NNEG) return;
    const float l0 = logits[2 * (NPOS + i)], l1 = logits[2 * (NPOS + i) + 1];
    float mx = fmaxf(l0, l1);
    float lse = mx + logf(expf(l0 - mx) + expf(l1 - mx));
    float p = expf(l0 - lse);                 // bg prob in [0,1]
    int bin = (int)(p * (float)NB);
    bin = bin < 0 ? 0 : (bin > NB - 1 ? NB - 1 : bin);
    atomicAdd(&hist[bin], 1);
}

__global__ void neg_scan_kernel(const int* __restrict__ hist, int* __restrict__ cut) {
    if (blockIdx.x == 0 && threadIdx.x == 0) {
        int cum = 0, c = NB - 1;
        for (int b = 0; b < NB; ++b) {
            cum += hist[b];
            if (cum >= NPOS * 20) { c = b; break; }
        }
        *cut = c;
    }
}

__global__ __launch_bounds__(256) void neg_sum_kernel(
    const float* __restrict__ logits, const int* __restrict__ cut,
    float* __restrict__ nsum, float* __restrict__ ncnt)
{
    __shared__ float red[256];
    const int cutb = *cut;
    int i = blockIdx.x * 256 + threadIdx.x;
    float s = 0.0f, c = 0.0f;
    if (i < NNEG) {
        const float l0 = logits[2 * (NPOS + i)], l1 = logits[2 * (NPOS + i) + 1];
        float mx = fmaxf(l0, l1);
        float lse = mx + logf(expf(l0 - mx) + expf(l1 - mx));
        float lp0 = l0 - lse;
        float p = expf(lp0);
        int bin = (int)(p * (float)NB);
        bin = bin < 0 ? 0 : (bin > NB - 1 ? NB - 1 : bin);
        if (bin <= cutb) { s = -lp0; c = 1.0f; }
    }
    float ts = block_reduce(s, red);
    __syncthreads();
    float tc = block_reduce(c, red);
    if (threadIdx.x == 0) { nsum[blockIdx.x] = ts; ncnt[blockIdx.x] = tc; }
}

__global__ __launch_bounds__(256) void final_kernel(
    const float* __restrict__ wsf, const float* __restrict__ nsum,
    const float* __restrict__ ncnt, float* __restrict__ out)
{
    __shared__ float red[256];
    int t = threadIdx.x;
    float s = 0.0f, c = 0.0f;
    for (int i = t; i < NEG_BLOCKS; i += 256) { s += nsum[i]; c += ncnt[i]; }
    float ts = block_reduce(s, red);
    __syncthreads();
    float tc = block_reduce(c, red);
    if (t == 0) {
        float neg_mean = ts / fmaxf(tc, 1.0f);
        out[POOL_OUT] = 0.5f * (wsf[WS_POS] + neg_mean);
    }
}

// ---------------------------------------------------------------------------
extern "C" void kernel_launch(void* const* d_in, const int* in_sizes, int n_in,
                              void* d_out, int out_size, void* d_ws, size_t ws_size,
                              hipStream_t stream) {
    const float* f0     = (const float*)d_in[0];
    const float* f1     = (const float*)d_in[1];
    const float* f2     = (const float*)d_in[2];
    const float* f3     = (const float*)d_in[3];
    const float* rois   = (const float*)d_in[4];
    const int*   am     = (const int*)d_in[5];
    const float* logits = (const float*)d_in[6];
    const float* pred   = (const float*)d_in[7];
    const float* targ   = (const float*)d_in[8];
    float* out = (float*)d_out;
    float* wsf = (float*)d_ws;
    int*   wsi = (int*)d_ws;
    (void)in_sizes; (void)n_in; (void)out_size; (void)ws_size;

    zero_hist_kernel<<<(NB + 255) / 256, 256, 0, stream>>>(wsi + WS_HIST);

    roialign_wmma_kernel<<<N_ROIS, 256, 0, stream>>>(f0, f1, f2, f3, rois, out);

    pos_bbox_kernel<<<1, 256, 0, stream>>>(am, logits, pred, targ, wsf, out);

    neg_hist_kernel<<<NEG_BLOCKS, 256, 0, stream>>>(logits, wsi + WS_HIST);
    neg_scan_kernel<<<1, 64, 0, stream>>>(wsi + WS_HIST, wsi + WS_CUT);
    neg_sum_kernel<<<NEG_BLOCKS, 256, 0, stream>>>(logits, wsi + WS_CUT,
                                                   wsf + WS_NSUM, wsf + WS_NCNT);
    final_kernel<<<1, 256, 0, stream>>>(wsf, wsf + WS_NSUM, wsf + WS_NCNT, out);
}